// BBPMMemory_80307298500956
// MI455X (gfx1250) — compile-verified
//
#include <hip/hip_runtime.h>
#include <hip/hip_bf16.h>
#include <math.h>

// ---------------------------------------------------------------------------
// BBPM memory: normalize + scatter-add (atomics) then count-normalized
// gather-mean.  Pure memory-system workload; tuned for gfx1250 wave32.
//   values   : [T, 64] f32        (d_in[0])
//   addresses: [T, 8]  int32      (d_in[1])  (JAX x64 off -> int32)
//   memory   : [D, 64] f32        (d_in[2])  (read-only; deltas go to d_ws)
//   counts   : [D, 1]  f32        (d_in[3])
//   out      : [T, 64] f32
// Workspace layout: [ delta_mem : D*64 f32 ][ delta_cnt : D f32 ]
// Only the <=T*K rows named in `addresses` are zeroed/used each call, so no
// 260MB blanket memset is needed (saves ~126MB of HBM writes per call).
// ---------------------------------------------------------------------------

#define KADDR 8
#define DIMV  64   // 64 f32 per row = 256B = one wave32 x float2, fully coalesced

#if defined(__has_builtin)
# if __has_builtin(__builtin_amdgcn_global_load_async_to_lds_b64)
#  define HAVE_ASYNC_LDS 1
# endif
# if __has_builtin(__builtin_amdgcn_s_wait_asynccnt)
#  define HAVE_WAIT_ASYNC 1
# endif
#endif

#define AS1 __attribute__((address_space(1)))
#define AS3 __attribute__((address_space(3)))
typedef __attribute__((ext_vector_type(2))) int   v2i;  // b64 payload type
typedef __attribute__((ext_vector_type(2))) float v2f;  // NT-friendly float2

__device__ __forceinline__ float wave32_reduce_sum(float x) {
  // 5-step butterfly across the wave32 (lowers to LDS permute path)
  #pragma unroll
  for (int off = 16; off > 0; off >>= 1) x += __shfl_xor(x, off, 32);
  return x;
}

// Load all 8 row addresses with two b128 loads -> one wait, 8 pointers ready.
__device__ __forceinline__ void load_addr8(const int* __restrict__ a,
                                           size_t (&ak)[KADDR]) {
  const int4 aA = reinterpret_cast<const int4*>(a)[0];
  const int4 aB = reinterpret_cast<const int4*>(a)[1];
  ak[0] = (size_t)aA.x; ak[1] = (size_t)aA.y;
  ak[2] = (size_t)aA.z; ak[3] = (size_t)aA.w;
  ak[4] = (size_t)aB.x; ak[5] = (size_t)aB.y;
  ak[6] = (size_t)aB.z; ak[7] = (size_t)aB.w;
}

// --- Phase 0: zero exactly the delta rows/counters this call will touch ---
__global__ __launch_bounds__(256) void bbpm_zero(
    const int* __restrict__ addrs,
    float* __restrict__ dmem, float* __restrict__ dcnt, int T) {
  const int lane = threadIdx.x & 31;
  const int wv   = threadIdx.x >> 5;          // 8 waves / block
  const int t    = blockIdx.x * 8 + wv;       // one wave per value row
  if (t >= T) return;
  const int* a = addrs + (size_t)t * KADDR;
  size_t ak[KADDR];
  load_addr8(a, ak);
  const v2f z = {0.0f, 0.0f};
  #pragma unroll
  for (int k = 0; k < KADDR; ++k)             // 8 b64 stores, back-to-back
    reinterpret_cast<v2f*>(dmem + ak[k] * DIMV)[lane] = z;  // duplicates benign
  if (lane < KADDR)                           // one predicated op: lane k -> slot k
    dcnt[(size_t)a[lane]] = 0.0f;
}

// --- Phase 1: l2-normalize each value row, scatter-add into delta workspace ---
__global__ __launch_bounds__(256) void bbpm_scatter(
    const float* __restrict__ values, const int* __restrict__ addrs,
    float* __restrict__ dmem, float* __restrict__ dcnt, int T) {
  const int lane = threadIdx.x & 31;
  const int wv   = threadIdx.x >> 5;
  const int t    = blockIdx.x * 8 + wv;
  if (t >= T) return;

  // values are streamed exactly once -> non-temporal load
  const v2f v = __builtin_nontemporal_load(
      reinterpret_cast<const v2f*>(values + (size_t)t * DIMV) + lane);

  const int* a = addrs + (size_t)t * KADDR;
  size_t ak[KADDR];
  load_addr8(a, ak);

  const float ss = wave32_reduce_sum(v.x * v.x + v.y * v.y);
  const float sc = 1.0f / (sqrtf(ss) + 1e-8f);
  const float sx = v.x * sc, sy = v.y * sc;

  #pragma unroll
  for (int k = 0; k < KADDR; ++k) {           // 16 f32 atomics, back-to-back
    float* dst = dmem + ak[k] * DIMV + lane * 2;
    atomicAdd(dst,     sx);                   // global_atomic_add_f32, L2-resident
    atomicAdd(dst + 1, sy);
  }
  if (lane < KADDR)                           // one predicated count atomic
    atomicAdd(dcnt + (size_t)a[lane], 1.0f);
}

// --- Phase 2: gather K rows (base + delta), count-normalize, mean over K ---
__global__ __launch_bounds__(256) void bbpm_gather(
    const int* __restrict__ addrs,
    const float* __restrict__ mem_in, const float* __restrict__ cnt_in,
    const float* __restrict__ dmem,   const float* __restrict__ dcnt,
    float* __restrict__ out, int T) {
  const int lane = threadIdx.x & 31;
  const int wv   = threadIdx.x >> 5;
  const int t    = blockIdx.x * 8 + wv;
#if defined(HAVE_ASYNC_LDS)
  __shared__ float lds[8 * 2 * KADDR * DIMV]; // 8 waves * 16 rows * 64 f32 = 32KB
  float* myl = lds + wv * (2 * KADDR * DIMV);
#endif
  if (t >= T) return;

  const int* a = addrs + (size_t)t * KADDR;
  size_t ak[KADDR];
  load_addr8(a, ak);

#if defined(HAVE_ASYNC_LDS)
  // Fire all 16 gathered rows down the async global->LDS pipe (ASYNCcnt),
  // overlap with the count loads below, then s_wait_asynccnt before use.
  #pragma unroll
  for (int k = 0; k < KADDR; ++k) {
    float* g0 = const_cast<float*>(mem_in) + ak[k] * DIMV + lane * 2;
    float* g1 = const_cast<float*>(dmem)   + ak[k] * DIMV + lane * 2;
    __builtin_amdgcn_global_load_async_to_lds_b64(
        (AS1 v2i*)g0, (AS3 v2i*)(myl + k * DIMV + lane * 2), 0, 0);
    __builtin_amdgcn_global_load_async_to_lds_b64(
        (AS1 v2i*)g1, (AS3 v2i*)(myl + (KADDR + k) * DIMV + lane * 2), 0, 0);
  }
#endif

  float inv[KADDR];
  #pragma unroll
  for (int k = 0; k < KADDR; ++k) {
    const float c = cnt_in[ak[k]] + dcnt[ak[k]];
    inv[k] = (1.0f / (float)KADDR) / fmaxf(c, 1.0f);
  }

  float accx = 0.0f, accy = 0.0f;
#if defined(HAVE_ASYNC_LDS)
# if defined(HAVE_WAIT_ASYNC)
  __builtin_amdgcn_s_wait_asynccnt(0);
# else
  asm volatile("s_wait_asynccnt 0" ::: "memory");
# endif
  #pragma unroll
  for (int k = 0; k < KADDR; ++k) {
    const float gx = myl[k * DIMV + lane * 2]     + myl[(KADDR + k) * DIMV + lane * 2];
    const float gy = myl[k * DIMV + lane * 2 + 1] + myl[(KADDR + k) * DIMV + lane * 2 + 1];
    accx = fmaf(gx, inv[k], accx);
    accy = fmaf(gy, inv[k], accy);
  }
#else
  #pragma unroll
  for (int k = 0; k < KADDR; ++k) {
    const v2f g0 = reinterpret_cast<const v2f*>(mem_in + ak[k] * DIMV)[lane];
    const v2f g1 = reinterpret_cast<const v2f*>(dmem   + ak[k] * DIMV)[lane];
    accx = fmaf(g0.x + g1.x, inv[k], accx);
    accy = fmaf(g0.y + g1.y, inv[k], accy);
  }
#endif

  // out is written exactly once -> non-temporal store
  v2f o; o.x = accx; o.y = accy;
  __builtin_nontemporal_store(
      o, reinterpret_cast<v2f*>(out + (size_t)t * DIMV) + lane);
}

extern "C" void kernel_launch(void* const* d_in, const int* in_sizes, int n_in,
                              void* d_out, int out_size, void* d_ws, size_t ws_size,
                              hipStream_t stream) {
  const float* values = (const float*)d_in[0];
  const int*   addrs  = (const int*)d_in[1];
  const float* mem_in = (const float*)d_in[2];
  const float* cnt_in = (const float*)d_in[3];

  const int    T = in_sizes[0] / DIMV;          // 65536
  const size_t D = (size_t)in_sizes[2] / DIMV;  // 1,048,576

  float* dmem = (float*)d_ws;                   // D*64 f32 delta table
  float* dcnt = dmem + D * DIMV;                // D f32 delta counts

  const dim3 blk(256);                          // 8 wave32 per block
  const dim3 grd((unsigned)((T + 7) / 8));
  hipLaunchKernelGGL(bbpm_zero, grd, blk, 0, stream,
                     addrs, dmem, dcnt, T);
  hipLaunchKernelGGL(bbpm_scatter, grd, blk, 0, stream,
                     values, addrs, dmem, dcnt, T);
  hipLaunchKernelGGL(bbpm_gather, grd, blk, 0, stream,
                     addrs, mem_in, cnt_in, dmem, dcnt, (float*)d_out, T);
}